// QuantumLayer_17248588661513
// MI455X (gfx1250) — compile-verified
//
#include <hip/hip_runtime.h>
#include <math.h>

#define BN 16384          // batch
#define DD 512            // feature dim
#define NQ 8              // data qubits
#define NW 9              // wires (incl. ancilla)
#define NL 4              // layers
#define LN_EPS 1e-5f

typedef __attribute__((ext_vector_type(2))) float v2f;
typedef __attribute__((ext_vector_type(8))) float v8f;

struct cpx { float re, im; };
__device__ __forceinline__ cpx cmul(cpx a, cpx b) {
    return { a.re * b.re - a.im * b.im, a.re * b.im + a.im * b.re };
}
__device__ __forceinline__ cpx cadd(cpx a, cpx b) { return { a.re + b.re, a.im + b.im }; }

// ---------------------------------------------------------------------------
// K1: angles = LN(tanh(x @ w_in^T + b_in)) * gamma + beta + timestep
// one wave per row; w_in staged to LDS as [k][j]
// ---------------------------------------------------------------------------
__global__ __launch_bounds__(256) void k1_angles(
    const float* __restrict__ x, const float* __restrict__ ts,
    const float* __restrict__ w_in, const float* __restrict__ b_in,
    const float* __restrict__ g, const float* __restrict__ beta,
    float* __restrict__ angles)
{
    __shared__ float wl[DD * NQ];   // wl[k*8 + j] = w_in[j][k]
    int tid = threadIdx.x;
    for (int e = tid; e < DD * NQ; e += 256) {
        int j = e & 7, k = e >> 3;
        wl[e] = w_in[j * DD + k];
    }
    __syncthreads();

    int wv = tid >> 5, lane = tid & 31;
    for (int row = blockIdx.x * 8 + wv; row < BN; row += gridDim.x * 8) {
        float acc[NQ];
        #pragma unroll
        for (int j = 0; j < NQ; ++j) acc[j] = 0.f;
        const float* xr = x + (size_t)row * DD;
        for (int k = lane; k < DD; k += 32) {
            float xv = xr[k];
            const float* wk = &wl[k * NQ];
            #pragma unroll
            for (int j = 0; j < NQ; ++j) acc[j] += xv * wk[j];
        }
        #pragma unroll
        for (int j = 0; j < NQ; ++j)
            for (int off = 16; off; off >>= 1) acc[j] += __shfl_xor(acc[j], off, 32);
        if (lane == 0) {
            float h[NQ], mu = 0.f;
            #pragma unroll
            for (int j = 0; j < NQ; ++j) { h[j] = tanhf(acc[j] + b_in[j]); mu += h[j]; }
            mu *= (1.f / NQ);
            float var = 0.f;
            #pragma unroll
            for (int j = 0; j < NQ; ++j) { float d = h[j] - mu; var += d * d; }
            var *= (1.f / NQ);
            float inv = rsqrtf(var + LN_EPS);
            float t0 = ts[0];
            #pragma unroll
            for (int j = 0; j < NQ; ++j)
                angles[(size_t)row * NQ + j] = (h[j] - mu) * inv * g[j] + beta[j] + t0;
        }
    }
}

// ---------------------------------------------------------------------------
// K2: product state psi0 per sample. u_i = RZ(qw[1,i]) RX(qw[0,i]) RZ(a) RX(a) |0>
// Realified K-major (rows 0..255 = Re, 256..511 = Im), stored PAIR-INTERLEAVED:
//   element (k,n) lives at psi0[((k>>1)*BN + n)*2 + (k&1)]
// so a WMMA B-fragment lane load (rows k+2h, k+2h+1) is one b64.
// ---------------------------------------------------------------------------
__global__ __launch_bounds__(256) void k2_psi0(
    const float* __restrict__ angles, const float* __restrict__ qw,
    float* __restrict__ psi0)
{
    __shared__ float lbuf[512 * 32];     // [k][s] 64 KB
    int t = threadIdx.x;
    int s = t & 31;           // sample within block
    int p = t >> 5;           // amplitude slice 0..7
    int n0 = blockIdx.x * 32;
    int n = n0 + s;

    cpx u[NQ][2];
    #pragma unroll
    for (int i = 0; i < NQ; ++i) {
        float w0 = qw[0 * NW + i], w1 = qw[1 * NW + i];
        float c0 = cosf(0.5f * w0), s0 = sinf(0.5f * w0);
        float c1 = cosf(0.5f * w1), s1 = sinf(0.5f * w1);
        cpx em = { c1, -s1 }, ep = { c1, s1 };
        cpx M00 = cmul(em, { c0, 0.f });
        cpx M01 = cmul(em, { 0.f, -s0 });
        cpx M10 = cmul(ep, { 0.f, -s0 });
        cpx M11 = cmul(ep, { c0, 0.f });
        float a = angles[(size_t)n * NQ + i];
        float ca = cosf(0.5f * a), sa = sinf(0.5f * a);
        cpx v0 = { ca * ca, -ca * sa };   // e^{-ia/2} * cos(a/2)
        cpx v1 = { sa * sa, -ca * sa };   // e^{+ia/2} * (-i sin(a/2))
        u[i][0] = cadd(cmul(M00, v0), cmul(M01, v1));
        u[i][1] = cadd(cmul(M10, v0), cmul(M11, v1));
    }
    for (int j = 0; j < 32; ++j) {
        int c = p * 32 + j;                      // reduced basis index, wire w -> bit (7-w)
        cpx amp = { 1.f, 0.f };
        #pragma unroll
        for (int w = 0; w < NQ; ++w) amp = cmul(amp, u[w][(c >> (7 - w)) & 1]);
        lbuf[c * 32 + s]         = amp.re;
        lbuf[(c + 256) * 32 + s] = amp.im;
    }
    __syncthreads();
    // coalesced paired write-out: one float2 per (k-pair, sample)
    for (int e = t; e < 256 * 32; e += 256) {
        int kp = e >> 5, ss = e & 31;
        float2 v = make_float2(lbuf[(2 * kp) * 32 + ss], lbuf[(2 * kp + 1) * 32 + ss]);
        *reinterpret_cast<float2*>(psi0 + ((size_t)kp * BN + n0 + ss) * 2) = v;
    }
}

// ---------------------------------------------------------------------------
// K3: build realified entangling unitary A (1024 x 512).
// One wave simulates one basis column (4 KB LDS state), 78 fixed gates.
// Full-state index k: wire w <-> bit (8-w); ancilla = bit 0.
// ---------------------------------------------------------------------------
__device__ __forceinline__ void g_cnot(float2* st, int lane, int cb, int tb) {
    for (int k = lane; k < 512; k += 32)
        if ((k & cb) && !(k & tb)) {
            float2 a = st[k], b = st[k | tb];
            st[k] = b; st[k | tb] = a;
        }
}
__device__ __forceinline__ void g_ry(float2* st, int lane, int wb, float c, float s) {
    for (int k = lane; k < 512; k += 32)
        if (!(k & wb)) {
            float2 a = st[k], b = st[k | wb];
            st[k]      = make_float2(c * a.x - s * b.x, c * a.y - s * b.y);
            st[k | wb] = make_float2(s * a.x + c * b.x, s * a.y + c * b.y);
        }
}
__device__ __forceinline__ void g_rz(float2* st, int lane, int wb, float c, float s) {
    for (int k = lane; k < 512; k += 32) {
        float2 a = st[k];
        float sg = (k & wb) ? s : -s;      // diag(e^{-it/2}, e^{+it/2})
        st[k] = make_float2(c * a.x - sg * a.y, c * a.y + sg * a.x);
    }
}

__global__ __launch_bounds__(256) void k3_uent(
    const float* __restrict__ qw, float* __restrict__ A)
{
    __shared__ float2 st_all[8][512];   // 32 KB
    int tid = threadIdx.x, wv = tid >> 5, lane = tid & 31;
    float2* st = st_all[wv];
    int col = blockIdx.x * 8 + wv;      // reduced input basis index (0..255)

    for (int k = lane; k < 512; k += 32) st[k] = make_float2(0.f, 0.f);
    if (lane == 0) st[col << 1] = make_float2(1.f, 0.f);  // ancilla bit = 0

    for (int l = 1; l < NL; ++l) {
        for (int i = 0; i < NQ; ++i) {
            int cb = 1 << (8 - i);
            int tb = 1 << (8 - ((i + 1) & 7));
            g_cnot(st, lane, cb, tb);
            float th = qw[l * NW + i];
            float c = cosf(0.5f * th), s = sinf(0.5f * th);
            g_ry(st, lane, cb, c, s);
            g_rz(st, lane, cb, c, s);
        }
        g_cnot(st, lane, 1 << 1, 1);    // control wire 7, target ancilla (wire 8)
        float th = qw[l * NW + 8];
        g_ry(st, lane, 1, cosf(0.5f * th), sinf(0.5f * th));
    }

    // realified A: [Re | -Im ; Im | Re], row-major [1024][512]
    for (int m = lane; m < 512; m += 32) {
        float2 v = st[m];
        A[(size_t)m * 512 + col]               =  v.x;
        A[(size_t)m * 512 + col + 256]         = -v.y;
        A[(size_t)(m + 512) * 512 + col]       =  v.y;
        A[(size_t)(m + 512) * 512 + col + 256] =  v.x;
    }
}

// ---------------------------------------------------------------------------
// K4: psi = A (1024x512) @ psi0 (512xBN) using V_WMMA_F32_16X16X4_F32.
// Each wave computes a 64x16 tile (4 accumulators): one B fragment (b64 from
// the pair-interleaved psi0) feeds 4 WMMAs -> ~1.25 VMEM per WMMA.
// Fragment layouts per ISA 7.12.2:
//   A[v][lane]: M = lane&15, K = v + 2*(lane>=16)
//   B[v][lane]: N = lane&15, K = v + 2*(lane>=16)
//   D[v][lane]: M = v + 8*(lane>=16), N = lane&15
// ---------------------------------------------------------------------------
__global__ __launch_bounds__(256) void k4_gemm(
    const float* __restrict__ A, const float* __restrict__ Bm,
    float* __restrict__ C)
{
    int tid = threadIdx.x, wv = tid >> 5, lane = tid & 31;
    int m0 = blockIdx.x * 64;
    int n0 = (blockIdx.y * 8 + wv) * 16;
    int half = lane >> 4;     // 0/1
    int l16  = lane & 15;

    const float* arow  = A + (size_t)(m0 + l16) * 512 + 2 * half;
    // B element (k,n) at Bm[((k>>1)*BN + n)*2 + (k&1)]; lane reads rows k+2h,k+2h+1
    const float* bbase = Bm + (size_t)(n0 + l16) * 2 + (size_t)half * 2 * BN;

    v8f acc[4];
    #pragma unroll
    for (int mm = 0; mm < 4; ++mm) acc[mm] = (v8f){0.f,0.f,0.f,0.f,0.f,0.f,0.f,0.f};

    #pragma unroll 2
    for (int k = 0; k < 512; k += 4) {
        v2f b  = *reinterpret_cast<const v2f*>(bbase + (size_t)k * BN);
        v2f a0 = *reinterpret_cast<const v2f*>(arow + k);
        v2f a1 = *reinterpret_cast<const v2f*>(arow + 16 * 512 + k);
        v2f a2 = *reinterpret_cast<const v2f*>(arow + 32 * 512 + k);
        v2f a3 = *reinterpret_cast<const v2f*>(arow + 48 * 512 + k);
        acc[0] = __builtin_amdgcn_wmma_f32_16x16x4_f32(false, a0, false, b, (short)0, acc[0], false, false);
        acc[1] = __builtin_amdgcn_wmma_f32_16x16x4_f32(false, a1, false, b, (short)0, acc[1], false, false);
        acc[2] = __builtin_amdgcn_wmma_f32_16x16x4_f32(false, a2, false, b, (short)0, acc[2], false, false);
        acc[3] = __builtin_amdgcn_wmma_f32_16x16x4_f32(false, a3, false, b, (short)0, acc[3], false, false);
    }

    #pragma unroll
    for (int mm = 0; mm < 4; ++mm) {
        float* crow = C + (size_t)(m0 + mm * 16 + 8 * half) * BN + n0 + l16;
        #pragma unroll
        for (int i = 0; i < 8; ++i) crow[(size_t)i * BN] = acc[mm][i];
    }
}

// ---------------------------------------------------------------------------
// K5: probs -> <Z_i> -> out = x + z @ w_out^T + b_out
// block handles 256 consecutive samples; all global traffic coalesced
// ---------------------------------------------------------------------------
__global__ __launch_bounds__(256) void k5_out(
    const float* __restrict__ psi, const float* __restrict__ x,
    const float* __restrict__ w_out, const float* __restrict__ b_out,
    float* __restrict__ out)
{
    __shared__ float zb[256][NQ];
    int t = threadIdx.x;
    int n = blockIdx.x * 256 + t;
    float z[NQ];
    #pragma unroll
    for (int i = 0; i < NQ; ++i) z[i] = 0.f;

    const float* pre = psi + n;
    for (int k = 0; k < 512; ++k) {
        float re = pre[(size_t)k * BN];
        float im = pre[(size_t)(k + 512) * BN];
        float p = re * re + im * im;
        #pragma unroll
        for (int i = 0; i < NQ; ++i) z[i] += ((k >> (8 - i)) & 1) ? -p : p;
    }
    #pragma unroll
    for (int i = 0; i < NQ; ++i) zb[t][i] = z[i];
    __syncthreads();

    size_t base = (size_t)blockIdx.x * 256 * DD;
    for (int e = t; e < 256 * DD; e += 256) {
        int nl = e >> 9, d = e & 511;
        float acc = x[base + e] + b_out[d];
        #pragma unroll
        for (int i = 0; i < NQ; ++i) acc += zb[nl][i] * w_out[d * NQ + i];
        out[base + e] = acc;
    }
}

// ---------------------------------------------------------------------------
extern "C" void kernel_launch(void* const* d_in, const int* in_sizes, int n_in,
                              void* d_out, int out_size, void* d_ws, size_t ws_size,
                              hipStream_t stream) {
    (void)in_sizes; (void)n_in; (void)out_size; (void)ws_size;
    const float* x        = (const float*)d_in[0];
    const float* ts       = (const float*)d_in[1];
    const float* w_in     = (const float*)d_in[2];
    const float* b_in     = (const float*)d_in[3];
    const float* ln_gamma = (const float*)d_in[4];
    const float* ln_beta  = (const float*)d_in[5];
    const float* qw       = (const float*)d_in[6];
    const float* w_out    = (const float*)d_in[7];
    const float* b_out    = (const float*)d_in[8];
    float* out = (float*)d_out;

    float* ws = (float*)d_ws;
    float* angles = ws;                                  // BN*8        (512 KB)
    float* psi0   = angles + (size_t)BN * NQ;            // 512*BN      (32 MB, pair-interleaved)
    float* Aent   = psi0 + (size_t)512 * BN;             // 1024*512    (2 MB)
    float* psiOut = Aent + (size_t)1024 * 512;           // 1024*BN     (64 MB)

    k1_angles<<<256, 256, 0, stream>>>(x, ts, w_in, b_in, ln_gamma, ln_beta, angles);
    k2_psi0 <<<BN / 32, 256, 0, stream>>>(angles, qw, psi0);
    k3_uent <<<32, 256, 0, stream>>>(qw, Aent);
    k4_gemm <<<dim3(1024 / 64, BN / 128), 256, 0, stream>>>(Aent, psi0, psiOut);
    k5_out  <<<BN / 256, 256, 0, stream>>>(psiOut, x, w_out, b_out, out);
}